// MultiScaleRetention3D_10960756539572
// MI455X (gfx1250) — compile-verified
//
#include <hip/hip_runtime.h>
#include <math.h>

// Shapes: B=1, C=64, D=H=W=16, HEADS=4, HD=16, P = D*H*W = 4096
#define NPOS 4096
#define NC   64

typedef __attribute__((ext_vector_type(2))) float v2f;
typedef __attribute__((ext_vector_type(8))) float v8f;

__device__ __forceinline__ v8f wmma4(v2f a, v2f b, v8f c) {
  // D(16x16,f32) += A(16x4,f32) * B(4x16,f32)  — full-precision CDNA5 WMMA
  return __builtin_amdgcn_wmma_f32_16x16x4_f32(false, a, false, b, (short)0, c,
                                               false, false);
}

__device__ __forceinline__ int iabs(int v) { return v < 0 ? -v : v; }

// ---------------------------------------------------------------------------
// Spatial conv: 1x3x3, pad (0,1,1).  x[64,4096] -> y[64,4096]
// ---------------------------------------------------------------------------
__global__ void conv_sp_kernel(const float* __restrict__ x,
                               const float* __restrict__ w,
                               const float* __restrict__ b,
                               float* __restrict__ y) {
  int idx = blockIdx.x * blockDim.x + threadIdx.x;  // 64*4096 threads
  int oc = idx >> 12;
  int p = idx & 4095;
  int d = p >> 8, h = (p >> 4) & 15, wx = p & 15;
  float acc = b[oc];
  for (int ic = 0; ic < NC; ++ic) {
    const float* wp = w + (oc * NC + ic) * 9;
    const float* xp = x + ic * NPOS + d * 256;
#pragma unroll
    for (int dh = -1; dh <= 1; ++dh) {
      int hh = h + dh;
      if (hh < 0 || hh > 15) continue;
#pragma unroll
      for (int dw = -1; dw <= 1; ++dw) {
        int ww = wx + dw;
        if (ww < 0 || ww > 15) continue;
        acc += wp[(dh + 1) * 3 + (dw + 1)] * xp[hh * 16 + ww];
      }
    }
  }
  y[idx] = acc;
}

// ---------------------------------------------------------------------------
// Spectral conv: 3x1x1, pad (1,0,0).  x[64,4096] -> y[64,4096]
// ---------------------------------------------------------------------------
__global__ void conv_spec_kernel(const float* __restrict__ x,
                                 const float* __restrict__ w,
                                 const float* __restrict__ b,
                                 float* __restrict__ y) {
  int idx = blockIdx.x * blockDim.x + threadIdx.x;
  int oc = idx >> 12;
  int p = idx & 4095;
  int d = p >> 8, hw = p & 255;
  float acc = b[oc];
  for (int ic = 0; ic < NC; ++ic) {
    const float* wp = w + (oc * NC + ic) * 3;
    const float* xp = x + ic * NPOS + hw;
#pragma unroll
    for (int dd = -1; dd <= 1; ++dd) {
      int ddn = d + dd;
      if (ddn < 0 || ddn > 15) continue;
      acc += wp[dd + 1] * xp[ddn * 256];
    }
  }
  y[idx] = acc;
}

// ---------------------------------------------------------------------------
// WMMA GEMM for 1x1x1 convs:  Y[M,4096] = Wm[M,64] * X[64,4096] + bias
// grid = (256 pos-tiles, M/16 oc-tiles), block = 32 (one wave)
// ---------------------------------------------------------------------------
__global__ __launch_bounds__(32) void gemm64_kernel(
    const float* __restrict__ Wm, const float* __restrict__ bias,
    const float* __restrict__ X, float* __restrict__ Y) {
  int l = threadIdx.x;
  int lo = l & 15;
  bool hi = l >= 16;
  int pt = blockIdx.x, ot = blockIdx.y;
  int pos = pt * 16 + lo;
  int oc = ot * 16 + lo;
  v8f acc = {};
#pragma unroll
  for (int kk = 0; kk < 16; ++kk) {
    int k0 = kk * 4 + (hi ? 2 : 0);
    int k1 = kk * 4 + (hi ? 3 : 1);
    v2f a = {Wm[oc * NC + k0], Wm[oc * NC + k1]};      // A[M=oc][K=ic]
    v2f b = {X[k0 * NPOS + pos], X[k1 * NPOS + pos]};  // B[K=ic][N=pos]
    acc = wmma4(a, b, acc);
  }
#pragma unroll
  for (int r = 0; r < 8; ++r) {
    int ocr = ot * 16 + r + (hi ? 8 : 0);
    Y[ocr * NPOS + pt * 16 + lo] = acc[r] + bias[ocr];
  }
}

// ---------------------------------------------------------------------------
// Attention: per (head, 16-query tile) loop over 256 key blocks of 16 keys.
//   S^T = K_blk(16x16hd) x Q(16hd x 16q)   (4x WMMA f32 16x16x4)
//   decay * S (decay via LDS g^e table), softmax over the local 16-key o axis
//   O    += V_blk(16d x 16o) x P^T(16o x 16q)  (4x WMMA)
// grid = (256 qtiles, 4 heads), block = 32
// ---------------------------------------------------------------------------
__global__ __launch_bounds__(32) void attn_kernel(
    const float* __restrict__ QKV, const float* __restrict__ gamma,
    float* __restrict__ O) {
  __shared__ float gp[48];  // gp[e] = g^e, e = |i-m|+|j-k|+|n-o| <= 45

  int l = threadIdx.x;
  int lo = l & 15;
  bool hi = l >= 16;
  int head = blockIdx.y;
  int qt = blockIdx.x;
  int q = qt * 16 + lo;
  int qd = q >> 8, qh = (q >> 4) & 15, qw = q & 15;

  const float* Q = QKV + (head * 16) * NPOS;
  const float* K = QKV + (64 + head * 16) * NPOS;
  const float* V = QKV + (128 + head * 16) * NPOS;

  float g = 1.0f / (1.0f + __expf(-gamma[0]));
  float lg = log2f(g);
  for (int e = l; e < 48; e += 32) gp[e] = exp2f((float)e * lg);
  __syncthreads();

  int cj = iabs(qh - qw);  // |j - k| : the reference's query-H vs query-W quirk

  // Q as B-operand (K=hd rows, N=query cols); fold in 1/sqrt(HD) = 0.25
  v2f qb[4];
#pragma unroll
  for (int kk = 0; kk < 4; ++kk) {
    int k0 = 4 * kk + (hi ? 2 : 0);
    int k1 = 4 * kk + (hi ? 3 : 1);
    qb[kk].x = Q[k0 * NPOS + q] * 0.25f;
    qb[kk].y = Q[k1 * NPOS + q] * 0.25f;
  }

  v8f out = {};
  for (int kb = 0; kb < 256; ++kb) {
    int m = kb >> 4, n = kb & 15;  // key D, key H
    int kpos = kb * 16 + lo;

    // keep L0 warm a few key-blocks ahead (gfx1250 global_prefetch_b8)
    int pkb = (kb + 8) & 255;
    __builtin_prefetch(&K[(hi ? 8 : 0) * NPOS + pkb * 16 + lo], 0, 1);
    __builtin_prefetch(&V[lo * NPOS + pkb * 16], 0, 1);

    // --- GEMM1: S^T[o, q] ---
    v8f s = {};
#pragma unroll
    for (int kk = 0; kk < 4; ++kk) {
      int k0 = 4 * kk + (hi ? 2 : 0);
      int k1 = 4 * kk + (hi ? 3 : 1);
      v2f a = {K[k0 * NPOS + kpos], K[k1 * NPOS + kpos]};  // A[M=o][K=hd]
      s = wmma4(a, qb[kk], s);
    }

    // --- decay (table lookup) + softmax over o ---
    int o0 = hi ? 8 : 0;
    float fb = gp[iabs(qd - m) + cj];  // g^(|i-m|+|j-k|), constant over o
    float p[8];
    float mx = -INFINITY;
#pragma unroll
    for (int r = 0; r < 8; ++r) {
      float val = s[r] * (fb * gp[iabs(n - (r + o0))]);
      p[r] = val;
      mx = fmaxf(mx, val);
    }
    mx = fmaxf(mx, __shfl_xor(mx, 16, 32));
    float sum = 0.0f;
#pragma unroll
    for (int r = 0; r < 8; ++r) {
      p[r] = __expf(p[r] - mx);
      sum += p[r];
    }
    sum += __shfl_xor(sum, 16, 32);
    float inv = 1.0f / sum;
#pragma unroll
    for (int r = 0; r < 8; ++r) p[r] *= inv;

    // --- rearrange P^T (C-layout, rows = o) into B-layout chunks ---
    float t[8];
#pragma unroll
    for (int r = 0; r < 8; ++r) t[r] = __shfl_xor(p[r], 16, 32);
    v2f pb[4];
    pb[0].x = hi ? t[2] : p[0];  pb[0].y = hi ? t[3] : p[1];   // rows 0..3
    pb[1].x = hi ? t[6] : p[4];  pb[1].y = hi ? t[7] : p[5];   // rows 4..7
    pb[2].x = hi ? p[2] : t[0];  pb[2].y = hi ? p[3] : t[1];   // rows 8..11
    pb[3].x = hi ? p[6] : t[4];  pb[3].y = hi ? p[7] : t[5];   // rows 12..15

    // --- GEMM2: O[d, q] += V[d, o] * P^T[o, q] ---
    const float* vrow = V + lo * NPOS + kb * 16;
#pragma unroll
    for (int kk = 0; kk < 4; ++kk) {
      int o_x = 4 * kk + (hi ? 2 : 0);
      int o_y = 4 * kk + (hi ? 3 : 1);
      v2f a = {vrow[o_x], vrow[o_y]};  // A[M=d][K=o]
      out = wmma4(a, pb[kk], out);
    }
  }

#pragma unroll
  for (int r = 0; r < 8; ++r) {
    int dch = head * 16 + r + (hi ? 8 : 0);
    O[dch * NPOS + qt * 16 + lo] = out[r];
  }
}

// ---------------------------------------------------------------------------
extern "C" void kernel_launch(void* const* d_in, const int* in_sizes, int n_in,
                              void* d_out, int out_size, void* d_ws,
                              size_t ws_size, hipStream_t stream) {
  (void)in_sizes; (void)n_in; (void)out_size; (void)ws_size;
  const float* x      = (const float*)d_in[0];
  const float* gamma  = (const float*)d_in[1];
  const float* w_sp   = (const float*)d_in[2];
  const float* b_sp   = (const float*)d_in[3];
  const float* w_spec = (const float*)d_in[4];
  const float* b_spec = (const float*)d_in[5];
  const float* w_qkv  = (const float*)d_in[6];
  const float* b_qkv  = (const float*)d_in[7];
  const float* w_proj = (const float*)d_in[8];
  const float* b_proj = (const float*)d_in[9];
  float* out = (float*)d_out;

  float* ws  = (float*)d_ws;
  float* x1  = ws;                      // [64, 4096]
  float* x2  = ws + 1 * NC * NPOS;      // [64, 4096]
  float* qkv = ws + 2 * NC * NPOS;      // [192, 4096]
  float* att = ws + 5 * NC * NPOS;      // [64, 4096]

  conv_sp_kernel<<<dim3(NC * NPOS / 256), dim3(256), 0, stream>>>(x, w_sp, b_sp, x1);
  conv_spec_kernel<<<dim3(NC * NPOS / 256), dim3(256), 0, stream>>>(x1, w_spec, b_spec, x2);
  gemm64_kernel<<<dim3(256, 12), dim3(32), 0, stream>>>(w_qkv, b_qkv, x2, qkv);
  attn_kernel<<<dim3(256, 4), dim3(32), 0, stream>>>(qkv, gamma, att);
  gemm64_kernel<<<dim3(256, 4), dim3(32), 0, stream>>>(w_proj, b_proj, att, out);
}